// EmbeddingShard_55705725829311
// MI455X (gfx1250) — compile-verified
//
#include <hip/hip_runtime.h>

// out = (gather(W, x) + sum_s b[s]) / 8   — pure HBM-bound gather.
// x: int32[16384], W: f32[8*6300, 4096], b: f32[8,4096], out: f32[16384,4096]

#define OUT_DIM        4096
#define SHARDS         8
#define THREADS        256
#define WAVES_PER_BLK  (THREADS / 32)            // 8
#define COLS_PER_WAVE  (OUT_DIM / WAVES_PER_BLK) // 512 floats = 2KB per wave
#define CHUNKS         (COLS_PER_WAVE / 128)     // 4 x b128-per-wave instructions
#define ROWS_PER_BLOCK 4

typedef __attribute__((ext_vector_type(4))) float f32x4;
typedef __attribute__((ext_vector_type(4))) int   i32x4;

#if defined(__has_builtin)
#if __has_builtin(__builtin_amdgcn_global_load_async_to_lds_b128)
#define HAVE_ASYNC_LDS 1
#endif
#endif

// s_wait_asynccnt with literal immediate; memory clobber stops the compiler
// from hoisting LDS reads above the wait.
#define WAIT_ASYNC(N) asm volatile("s_wait_asynccnt " #N ::: "memory")

// ---------------------------------------------------------------------------
// Kernel 1: bs[j] = 0.125f * sum_s b[s][j]   (4096 floats into d_ws)
// ---------------------------------------------------------------------------
__global__ void bias_sum_kernel(const float* __restrict__ b,
                                float* __restrict__ bs) {
    int j = blockIdx.x * blockDim.x + threadIdx.x;
    if (j < OUT_DIM) {
        float s = 0.0f;
#pragma unroll
        for (int k = 0; k < SHARDS; ++k)
            s += b[k * OUT_DIM + j];
        bs[j] = s * 0.125f;
    }
}

// ---------------------------------------------------------------------------
// Kernel 2: gathered rows via async global->LDS copy, double-buffered.
// Each wave owns a disjoint 512-float strip per row => no barriers, only
// per-wave s_wait_asynccnt. Output streamed with non-temporal b128 stores.
// ---------------------------------------------------------------------------
__global__ __launch_bounds__(THREADS)
void gather_kernel(const float* __restrict__ W,
                   const int*   __restrict__ x,
                   const float* __restrict__ bs,
                   float*       __restrict__ out,
                   int nRows) {
#if HAVE_ASYNC_LDS
    __shared__ __align__(16) float lds[2][OUT_DIM];   // 32 KB double buffer
#endif
    const int tid     = threadIdx.x;
    const int wave    = tid >> 5;
    const int lane    = tid & 31;
    const int colBase = wave * COLS_PER_WAVE;
    const long rowBase = (long)blockIdx.x * ROWS_PER_BLOCK;

    // Row indices for this block (clamped for safety on ragged tails).
    int rows[ROWS_PER_BLOCK];
#pragma unroll
    for (int r = 0; r < ROWS_PER_BLOCK; ++r) {
        long ri = rowBase + r;
        rows[r] = x[ri < nRows ? ri : (nRows - 1)];
    }

    // Bias strip for this wave, held in registers across all rows (already /8).
    f32x4 bsv[CHUNKS];
#pragma unroll
    for (int k = 0; k < CHUNKS; ++k)
        bsv[k] = *(const f32x4*)(bs + colBase + k * 128 + lane * 4);

#if HAVE_ASYNC_LDS
    // Issue one row's strip: 4 async b128 instructions per wave (32 lanes x 16B).
    auto issue = [&](int buf, int row) {
#pragma unroll
        for (int k = 0; k < CHUNKS; ++k) {
            const int off = colBase + k * 128 + lane * 4;
            float* g = const_cast<float*>(W + (size_t)row * OUT_DIM + off);
            __builtin_amdgcn_global_load_async_to_lds_b128(
                (__attribute__((address_space(1))) i32x4*)g,
                (__attribute__((address_space(3))) i32x4*)(&lds[buf][off]),
                0, 0);
        }
    };

    issue(0, rows[0]);
#pragma unroll
    for (int r = 0; r < ROWS_PER_BLOCK; ++r) {
        if (r + 1 < ROWS_PER_BLOCK)
            issue((r + 1) & 1, rows[r + 1]);
        // Async loads complete in order: <=CHUNKS outstanding means row r done.
        if (r + 1 < ROWS_PER_BLOCK) {
            WAIT_ASYNC(4);
        } else {
            WAIT_ASYNC(0);
        }
        if (rowBase + r < nRows) {
            float* dst = out + (size_t)(rowBase + r) * OUT_DIM + colBase;
#pragma unroll
            for (int k = 0; k < CHUNKS; ++k) {
                const int off = colBase + k * 128 + lane * 4;
                f32x4 v = *(const f32x4*)(&lds[r & 1][off]);
                f32x4 o = v * 0.125f + bsv[k];
                __builtin_nontemporal_store(o, (f32x4*)(dst + (k * 128 + lane * 4)));
            }
        }
    }
#else
    // Fallback: direct vectorized gather (no LDS staging).
#pragma unroll
    for (int r = 0; r < ROWS_PER_BLOCK; ++r) {
        if (rowBase + r >= nRows) break;
        const float* src = W + (size_t)rows[r] * OUT_DIM + colBase;
        float*       dst = out + (size_t)(rowBase + r) * OUT_DIM + colBase;
#pragma unroll
        for (int k = 0; k < CHUNKS; ++k) {
            const int off = k * 128 + lane * 4;
            f32x4 v = *(const f32x4*)(src + off);
            f32x4 o = v * 0.125f + bsv[k];
            __builtin_nontemporal_store(o, (f32x4*)(dst + off));
        }
    }
#endif
}

// ---------------------------------------------------------------------------
extern "C" void kernel_launch(void* const* d_in, const int* in_sizes, int n_in,
                              void* d_out, int out_size, void* d_ws, size_t ws_size,
                              hipStream_t stream) {
    const int*   x = (const int*)d_in[0];    // [16384] int32 indices
    const float* W = (const float*)d_in[1];  // [8*6300*4096] f32
    const float* b = (const float*)d_in[2];  // [8*4096] f32
    float* out = (float*)d_out;
    float* bs  = (float*)d_ws;               // 4096 f32 scratch (16 KB)

    bias_sum_kernel<<<(OUT_DIM + THREADS - 1) / THREADS, THREADS, 0, stream>>>(b, bs);

    const int nRows = in_sizes[0];           // 16384
    const int grid  = (nRows + ROWS_PER_BLOCK - 1) / ROWS_PER_BLOCK;
    gather_kernel<<<grid, THREADS, 0, stream>>>(W, x, bs, out, nRows);
}